// BaseHead_18777597018225
// MI455X (gfx1250) — compile-verified
//
#include <hip/hip_runtime.h>
#include <stdint.h>

#define NTHREADS 256
#define TMAX     8192

// Order-preserving float32 -> uint32 key (unsigned compare == float compare).
__device__ __forceinline__ uint32_t f2k(uint32_t x) {
    return (x & 0x80000000u) ? ~x : (x | 0x80000000u);
}
__device__ __forceinline__ float k2f(uint32_t u) {
    uint32_t x = (u & 0x80000000u) ? (u & 0x7FFFFFFFu) : ~u;
    return __uint_as_float(x);
}

// One workgroup (8 wave32s) per row. Async-DMA the valid prefix of the row
// into LDS, 4-pass (MSB->LSB) radix-select the k-th largest, then reduce.
__global__ __launch_bounds__(NTHREADS) void topk_mean_kernel(
    const float* __restrict__ logits,
    const int*   __restrict__ seq_len,
    float*       __restrict__ out,
    int T) {
    __shared__ uint32_t s_keys[TMAX];     // row data: floats, then keys (in place)
    __shared__ uint32_t s_hist[256];      // radix histogram
    __shared__ uint32_t s_scan[256];      // suffix-scan workspace
    __shared__ float    s_rf[NTHREADS];   // sum reduction
    __shared__ uint32_t s_ru[NTHREADS];   // count reduction
    __shared__ uint32_t s_bc[2];          // [0]=selected digit, [1]=remaining rank

    const int tid = threadIdx.x;
    const int row = blockIdx.x;
    const int s   = seq_len[row];         // 1..T
    const int k   = (s >> 4) + 1;         // k <= s always

    // ---- Stage 1: async HBM -> LDS copy of the valid prefix (16B/lane/op) ----
    const uint64_t gbase = (uint64_t)(uintptr_t)(logits + (size_t)row * (size_t)T);
    const uint32_t lbase = (uint32_t)(uintptr_t)(&s_keys[0]); // low 32 bits = LDS offset
    const int nchunks = (s + 3) >> 2;     // 4 floats per 16B chunk
    for (int c = tid; c < nchunks; c += NTHREADS) {
        const uint32_t off   = (uint32_t)c << 4;   // byte offset in row
        const uint32_t laddr = lbase + off;        // LDS byte address
        asm volatile("global_load_async_to_lds_b128 %0, %1, %2"
                     :: "v"(laddr), "v"(off), "s"(gbase) : "memory");
    }
    asm volatile("s_wait_asynccnt 0" ::: "memory");

    s_hist[tid] = 0u;
    if (tid == 0) s_bc[1] = (uint32_t)k;
    __syncthreads();   // async data landed (each wave drained its own ASYNCcnt)

    // ---- Stage 2: in-place key transform fused with the MSB histogram ----
    {
        uint4* kv = (uint4*)s_keys;
        const int nvec = s >> 2;                  // full 4-element chunks
        for (int c = tid; c < nvec; c += NTHREADS) {
            uint4 q = kv[c];
            q.x = f2k(q.x); q.y = f2k(q.y); q.z = f2k(q.z); q.w = f2k(q.w);
            kv[c] = q;
            atomicAdd(&s_hist[q.x >> 24], 1u);
            atomicAdd(&s_hist[q.y >> 24], 1u);
            atomicAdd(&s_hist[q.z >> 24], 1u);
            atomicAdd(&s_hist[q.w >> 24], 1u);
        }
        for (int e = (nvec << 2) + tid; e < s; e += NTHREADS) {  // tail (<=3)
            uint32_t key = f2k(s_keys[e]);
            s_keys[e] = key;
            atomicAdd(&s_hist[key >> 24], 1u);
        }
    }
    __syncthreads();

    // ---- Stage 3: radix select; per pass: histogram + parallel suffix scan ----
    uint32_t pref = 0u;
    for (int pass = 0; pass < 4; ++pass) {
        const int shift = 24 - 8 * pass;

        if (pass > 0) {                   // pass 0 histogram was fused above
            s_hist[tid] = 0u;
            __syncthreads();
            const uint32_t phi = pref >> (shift + 8);
            const uint4* kv = (const uint4*)s_keys;
            const int nvec = s >> 2;
            for (int c = tid; c < nvec; c += NTHREADS) {
                uint4 q = kv[c];
                if ((q.x >> (shift + 8)) == phi) atomicAdd(&s_hist[(q.x >> shift) & 0xFFu], 1u);
                if ((q.y >> (shift + 8)) == phi) atomicAdd(&s_hist[(q.y >> shift) & 0xFFu], 1u);
                if ((q.z >> (shift + 8)) == phi) atomicAdd(&s_hist[(q.z >> shift) & 0xFFu], 1u);
                if ((q.w >> (shift + 8)) == phi) atomicAdd(&s_hist[(q.w >> shift) & 0xFFu], 1u);
            }
            for (int e = (nvec << 2) + tid; e < s; e += NTHREADS) {
                uint32_t key = s_keys[e];
                if ((key >> (shift + 8)) == phi)
                    atomicAdd(&s_hist[(key >> shift) & 0xFFu], 1u);
            }
            __syncthreads();
        }

        const uint32_t need = s_bc[1];    // written before last barrier -> safe
        // Inclusive suffix scan: s_scan[t] = # keys with digit >= (255-t).
        s_scan[tid] = s_hist[255 - tid];
        __syncthreads();
        #pragma unroll
        for (int off = 1; off < 256; off <<= 1) {
            const uint32_t v = s_scan[tid] + ((tid >= off) ? s_scan[tid - off] : 0u);
            __syncthreads();
            s_scan[tid] = v;
            __syncthreads();
        }
        const uint32_t cum  = s_scan[tid];
        const uint32_t prev = tid ? s_scan[tid - 1] : 0u;
        if (cum >= need && prev < need) { // exactly one thread crosses the rank
            s_bc[0] = (uint32_t)(255 - tid);
            s_bc[1] = need - prev;
        }
        __syncthreads();
        pref |= s_bc[0] << shift;
    }
    const uint32_t kt = pref;             // k-th largest key value

    // ---- Stage 4: sum of strictly-greater elements + tie fill ----
    float    psum = 0.0f;
    uint32_t pcnt = 0u;
    {
        const uint4* kv = (const uint4*)s_keys;
        const int nvec = s >> 2;
        for (int c = tid; c < nvec; c += NTHREADS) {
            uint4 q = kv[c];
            if (q.x > kt) { psum += k2f(q.x); pcnt++; }
            if (q.y > kt) { psum += k2f(q.y); pcnt++; }
            if (q.z > kt) { psum += k2f(q.z); pcnt++; }
            if (q.w > kt) { psum += k2f(q.w); pcnt++; }
        }
        for (int e = (nvec << 2) + tid; e < s; e += NTHREADS) {
            uint32_t key = s_keys[e];
            if (key > kt) { psum += k2f(key); pcnt++; }
        }
    }
    s_rf[tid] = psum;
    s_ru[tid] = pcnt;
    __syncthreads();
    #pragma unroll
    for (int off = NTHREADS / 2; off > 0; off >>= 1) {
        if (tid < off) {
            s_rf[tid] += s_rf[tid + off];
            s_ru[tid] += s_ru[tid + off];
        }
        __syncthreads();
    }
    if (tid == 0) {
        float sum = s_rf[0];
        const int cgt = (int)s_ru[0];          // # strictly greater, < k
        sum += (float)(k - cgt) * k2f(kt);     // fill remaining ranks with ties
        out[row] = sum / (float)k;
    }
}

extern "C" void kernel_launch(void* const* d_in, const int* in_sizes, int n_in,
                              void* d_out, int out_size, void* d_ws, size_t ws_size,
                              hipStream_t stream) {
    const float* logits = (const float*)d_in[0];
    const int*   seq    = (const int*)d_in[1];
    float*       out    = (float*)d_out;

    const int B = in_sizes[1];            // 4096
    const int T = in_sizes[0] / B;        // 8192 (fits TMAX LDS row buffer)

    topk_mean_kernel<<<dim3(B), dim3(NTHREADS), 0, stream>>>(logits, seq, out, T);
}